// CausalSelfAttention_20684562498031
// MI455X (gfx1250) — compile-verified
//
#include <hip/hip_runtime.h>

typedef __attribute__((ext_vector_type(2))) float v2f;
typedef __attribute__((ext_vector_type(8))) float v8f;
typedef __attribute__((ext_vector_type(4))) int   v4i;

#define D_MODEL 1024
#define N_HEAD  16
#define HEAD_DIM 64
#define BATCH   4
#define SEQ     2048
#define BSROWS  (BATCH * SEQ)   // 8192

#if defined(__has_builtin)
#  if __has_builtin(__builtin_amdgcn_global_load_async_to_lds_b128)
#    define HAVE_ASYNC_LDS 1
#  endif
#endif

// 16-byte global -> LDS copy. CDNA5 path: GLOBAL_LOAD_ASYNC_TO_LDS_B128
// (ASYNCcnt-tracked, no VGPR round trip). Fallback: sync float4 copy.
// Builtin signature (from compiler diagnostic): (v4i AS1*, v4i AS3*, Ii, Ii).
__device__ __forceinline__ void async_copy16(const float* __restrict__ g, float* l) {
#ifdef HAVE_ASYNC_LDS
    typedef __attribute__((address_space(1))) v4i* gptr_v4i;
    typedef __attribute__((address_space(3))) v4i* lptr_v4i;
    __builtin_amdgcn_global_load_async_to_lds_b128((gptr_v4i)g, (lptr_v4i)l, 0, 0);
#else
    *(float4*)l = *(const float4*)g;
#endif
}

__device__ __forceinline__ void wait_async_le4() {
#ifdef HAVE_ASYNC_LDS
#  if __has_builtin(__builtin_amdgcn_s_wait_asynccnt)
    __builtin_amdgcn_s_wait_asynccnt(4);
#  else
    asm volatile("s_wait_asynccnt 0x4" ::: "memory");
#  endif
#endif
}
__device__ __forceinline__ void wait_async_0() {
#ifdef HAVE_ASYNC_LDS
#  if __has_builtin(__builtin_amdgcn_s_wait_asynccnt)
    __builtin_amdgcn_s_wait_asynccnt(0);
#  else
    asm volatile("s_wait_asynccnt 0x0" ::: "memory");
#  endif
#endif
}

__device__ __forceinline__ v8f wmma_f32(v2f a, v2f b, v8f c) {
    // D = A(16x4 f32) * B(4x16 f32) + C(16x16 f32), full fp32 tensor path
    return __builtin_amdgcn_wmma_f32_16x16x4_f32(
        /*neg_a=*/false, a, /*neg_b=*/false, b,
        /*c_mod=*/(short)0, c, /*reuse_a=*/false, /*reuse_b=*/false);
}

// ---------------------------------------------------------------------------
// GEMM: C[M,N] = A[M,K] @ W[K,N] + bias[N]   (fp32, WMMA 16x16x4)
// Block: 256 threads = 8 waves; tile 128(M) x 128(N); K chunks of 16,
// double-buffered LDS filled by async-to-LDS copies. Wave w: wm = w>>1 owns a
// 32-row strip, wn = w&1 owns a 64-col half -> 8 WMMAs per K-step.
// ---------------------------------------------------------------------------
__global__ __launch_bounds__(256)
void gemm_bias_f32(const float* __restrict__ A, const float* __restrict__ W,
                   const float* __restrict__ bias, float* __restrict__ C,
                   int M, int N, int Kd) {
    // A rows padded to 20 floats: 80B stride -> async b128 dests 16B-aligned,
    // v2f frag reads are 8B-aligned ds_load_b64, banks disperse.
    __shared__ __align__(16) float lA[2][128][20];
    __shared__ __align__(16) float lB[2][16][128];

    const int t    = threadIdx.x;
    const int lane = t & 31, wave = t >> 5;
    const int half = lane >> 4, ln = lane & 15;
    const int wm   = wave >> 1, wn = wave & 1;
    const int m0   = blockIdx.y * 128;
    const int n0   = blockIdx.x * 128;

    v8f acc[2][4] = {};

    // staging assignment: A chunk 128x16 = 512 float4, B chunk 16x128 = 512 float4
    const int aRow = t >> 2;                // 0..63 (+64 for second half)
    const int aQ   = (t & 3) * 4;           // 0,4,8,12
    const int bK   = t >> 5;                // 0..7 (+8 for second half)
    const int bCol = (t & 31) * 4;          // 0..124

    auto issueChunk = [&](int k0, int buf) {   // 4 async ops per thread
        async_copy16(A + (size_t)(m0 + aRow) * Kd + k0 + aQ,      &lA[buf][aRow][aQ]);
        async_copy16(A + (size_t)(m0 + aRow + 64) * Kd + k0 + aQ, &lA[buf][aRow + 64][aQ]);
        async_copy16(W + (size_t)(k0 + bK) * N + n0 + bCol,       &lB[buf][bK][bCol]);
        async_copy16(W + (size_t)(k0 + bK + 8) * N + n0 + bCol,   &lB[buf][bK + 8][bCol]);
    };
    auto compute = [&](int buf) {
        #pragma unroll
        for (int kk = 0; kk < 4; ++kk) {
            const int kb = kk * 4 + half * 2;   // A/B-frag K layout per ISA
            const v2f a0 = *(const v2f*)&lA[buf][wm * 32 + ln][kb];
            const v2f a1 = *(const v2f*)&lA[buf][wm * 32 + 16 + ln][kb];
            #pragma unroll
            for (int j = 0; j < 4; ++j) {
                const int col = wn * 64 + j * 16 + ln;
                v2f b;
                b.x = lB[buf][kb][col];
                b.y = lB[buf][kb + 1][col];
                acc[0][j] = wmma_f32(a0, b, acc[0][j]);
                acc[1][j] = wmma_f32(a1, b, acc[1][j]);
            }
        }
    };

    const int nK = Kd / 16;
    issueChunk(0, 0);
    for (int kc = 0; kc < nK; ++kc) {
        const int buf = kc & 1;
        if (kc + 1 < nK) {
            issueChunk((kc + 1) * 16, buf ^ 1);  // keep next chunk in flight
            wait_async_le4();                     // chunk kc landed (mine)
        } else {
            wait_async_0();
        }
        __syncthreads();                          // all waves' copies landed
        compute(buf);
        __syncthreads();                          // done reading buf
    }

    // epilogue: C/D layout -> VGPR r holds row r (lanes 0-15) / r+8 (16-31)
    #pragma unroll
    for (int s = 0; s < 2; ++s) {
        #pragma unroll
        for (int j = 0; j < 4; ++j) {
            const int col = n0 + wn * 64 + j * 16 + ln;
            const float bval = bias[col];
            #pragma unroll
            for (int r = 0; r < 8; ++r) {
                const int row = m0 + wm * 32 + s * 16 + half * 8 + r;
                C[(size_t)row * N + col] = acc[s][j][r] + bval;
            }
        }
    }
}

// ---------------------------------------------------------------------------
// Flash attention (causal, fp32 WMMA). Block: 128 threads = 4 waves,
// 64 query rows per block; each wave owns 16 rows. Key tiles of 64.
// K kept in natural [key][hd] layout: async-stageable AND contiguous
// b64 B-fragments for Q@K^T. P kept [qrow][key]: b64 A-frags for P@V.
// ---------------------------------------------------------------------------
__global__ __launch_bounds__(128)
void flash_attn_f32(const float* __restrict__ Qg, const float* __restrict__ Kg,
                    const float* __restrict__ Vg, float* __restrict__ Og) {
    __shared__ __align__(16) float sK[64][68];       // [key][hd], pad 68
    __shared__ __align__(16) float sV[64][64];       // [key][hd]
    __shared__ __align__(16) float sP[4][16][68];    // per wave: [qrow][key]

    const int t    = threadIdx.x;
    const int lane = t & 31, wave = t >> 5;
    const int half = lane >> 4, ln = lane & 15;
    const int qt   = blockIdx.x;         // query tile (64 rows)
    const int h    = blockIdx.y;
    const int b    = blockIdx.z;
    const int q0   = qt * 64;
    const size_t rowBase = (size_t)b * SEQ;
    const int hb   = h * HEAD_DIM;

    // Q fragments in registers for the whole block, pre-scaled by 1/sqrt(hd)
    const int qrow = q0 + wave * 16 + ln;
    const float* qp = Qg + (rowBase + qrow) * D_MODEL + hb;
    v2f qf[16];
    #pragma unroll
    for (int ks = 0; ks < 16; ++ks) {
        const int kb = ks * 4 + half * 2;
        qf[ks] = *(const v2f*)(qp + kb) * 0.125f;
    }

    float mrow[8], lrow[8], alpha[8];
    v8f oacc[4] = {};
    #pragma unroll
    for (int r = 0; r < 8; ++r) { mrow[r] = -__builtin_inff(); lrow[r] = 0.0f; }

    const int nkt = qt + 1;              // causal: keys up to q0+63
    for (int kt = 0; kt < nkt; ++kt) {
        const int k0 = kt * 64;
        __syncthreads();                 // everyone done with previous tile
        // stage K and V tiles via async-to-LDS (16 ops/thread)
        #pragma unroll
        for (int i = 0; i < 8; ++i) {
            const int row = i * 8 + (t >> 4);   // key index 0..63
            const int col = (t & 15) * 4;       // hd offset
            async_copy16(Kg + (rowBase + k0 + row) * D_MODEL + hb + col, &sK[row][col]);
            async_copy16(Vg + (rowBase + k0 + row) * D_MODEL + hb + col, &sV[row][col]);
        }
        wait_async_0();
        __syncthreads();

        // S = (Q*scale) @ K^T : B[k=hd][n=key] = sK[n][k] -> contiguous b64
        v8f sacc[4] = {};
        #pragma unroll
        for (int ks = 0; ks < 16; ++ks) {
            const int kb = ks * 4 + half * 2;
            const v2f a = qf[ks];
            #pragma unroll
            for (int j = 0; j < 4; ++j) {
                const v2f bf = *(const v2f*)&sK[j * 16 + ln][kb];
                sacc[j] = wmma_f32(a, bf, sacc[j]);
            }
        }

        // causal mask (elementwise in C/D layout)
        #pragma unroll
        for (int j = 0; j < 4; ++j) {
            const int cg = k0 + j * 16 + ln;
            #pragma unroll
            for (int r = 0; r < 8; ++r) {
                const int rg = q0 + wave * 16 + half * 8 + r;
                if (cg > rg) sacc[j][r] = -__builtin_inff();
            }
        }

        // online softmax: rows live across 16 lanes of each half-wave
        #pragma unroll
        for (int r = 0; r < 8; ++r) {
            float tm = fmaxf(fmaxf(sacc[0][r], sacc[1][r]),
                             fmaxf(sacc[2][r], sacc[3][r]));
            tm = fmaxf(tm, __shfl_xor(tm, 1, 32));
            tm = fmaxf(tm, __shfl_xor(tm, 2, 32));
            tm = fmaxf(tm, __shfl_xor(tm, 4, 32));
            tm = fmaxf(tm, __shfl_xor(tm, 8, 32));
            const float mn = fmaxf(mrow[r], tm);
            alpha[r] = __expf(mrow[r] - mn);
            mrow[r]  = mn;
        }
        #pragma unroll
        for (int r = 0; r < 8; ++r) {
            float rs = 0.0f;
            #pragma unroll
            for (int j = 0; j < 4; ++j) {
                const float p = __expf(sacc[j][r] - mrow[r]);
                sacc[j][r] = p;
                rs += p;
            }
            rs += __shfl_xor(rs, 1, 32);
            rs += __shfl_xor(rs, 2, 32);
            rs += __shfl_xor(rs, 4, 32);
            rs += __shfl_xor(rs, 8, 32);
            lrow[r] = lrow[r] * alpha[r] + rs;
            #pragma unroll
            for (int j = 0; j < 4; ++j) oacc[j][r] *= alpha[r];
        }

        // P -> LDS in [qrow][key] layout (conflict-free stores, b64 A-frags)
        #pragma unroll
        for (int j = 0; j < 4; ++j)
            #pragma unroll
            for (int r = 0; r < 8; ++r)
                sP[wave][half * 8 + r][j * 16 + ln] = sacc[j][r];

        // O += P @ V : A-frag = sP[ln][kb..kb+1] (b64), B = sV[kb][col]
        #pragma unroll
        for (int ks = 0; ks < 16; ++ks) {
            const int kb = ks * 4 + half * 2;
            const v2f a = *(const v2f*)&sP[wave][ln][kb];
            #pragma unroll
            for (int j = 0; j < 4; ++j) {
                const int col = j * 16 + ln;
                v2f bf;
                bf.x = sV[kb][col];
                bf.y = sV[kb + 1][col];
                oacc[j] = wmma_f32(a, bf, oacc[j]);
            }
        }
    }

    // normalize and write out in [B*S, D] layout (head-interleaved columns)
    #pragma unroll
    for (int r = 0; r < 8; ++r) {
        const float inv = 1.0f / lrow[r];
        const int rg = q0 + wave * 16 + half * 8 + r;
        float* op = Og + (rowBase + rg) * D_MODEL + hb;
        #pragma unroll
        for (int j = 0; j < 4; ++j)
            op[j * 16 + ln] = oacc[j][r] * inv;
    }
}

// ---------------------------------------------------------------------------
extern "C" void kernel_launch(void* const* d_in, const int* in_sizes, int n_in,
                              void* d_out, int out_size, void* d_ws, size_t ws_size,
                              hipStream_t stream) {
    const float* x   = (const float*)d_in[0];
    const float* w_q = (const float*)d_in[1];
    const float* b_q = (const float*)d_in[2];
    const float* w_k = (const float*)d_in[3];
    const float* b_k = (const float*)d_in[4];
    const float* w_v = (const float*)d_in[5];
    const float* b_v = (const float*)d_in[6];
    const float* w_o = (const float*)d_in[7];
    const float* b_o = (const float*)d_in[8];
    float* out = (float*)d_out;
    float* ws  = (float*)d_ws;

    const size_t NEL = (size_t)BSROWS * D_MODEL;   // 8M floats = 32 MB
    float* Qb = ws;
    float* Kb = ws + NEL;
    float* Vb = ws + 2 * NEL;
    float* Ob = ws + 3 * NEL;

    dim3 gG(D_MODEL / 128, BSROWS / 128);          // 8 x 64 blocks
    gemm_bias_f32<<<gG, 256, 0, stream>>>(x, w_q, b_q, Qb, BSROWS, D_MODEL, D_MODEL);
    gemm_bias_f32<<<gG, 256, 0, stream>>>(x, w_k, b_k, Kb, BSROWS, D_MODEL, D_MODEL);
    gemm_bias_f32<<<gG, 256, 0, stream>>>(x, w_v, b_v, Vb, BSROWS, D_MODEL, D_MODEL);

    dim3 gA(SEQ / 64, N_HEAD, BATCH);              // 32 x 16 x 4 blocks
    flash_attn_f32<<<gA, 128, 0, stream>>>(Qb, Kb, Vb, Ob);

    gemm_bias_f32<<<gG, 256, 0, stream>>>(Ob, w_o, b_o, out, BSROWS, D_MODEL, D_MODEL);
}